// DiffusionConv_63780264345945
// MI455X (gfx1250) — compile-verified
//
#include <hip/hip_runtime.h>

typedef float v2f __attribute__((ext_vector_type(2)));
typedef float v8f __attribute__((ext_vector_type(8)));

#define TT 12     // T
#define CC 32     // C
#define CT 384    // C*T per node

// out[n*CT + c*TT + t] = bias[c]
__global__ void init_out_kernel(float* __restrict__ out,
                                const float* __restrict__ bias, int total) {
  int i = blockIdx.x * blockDim.x + threadIdx.x;
  if (i < total) {
    int r = i % CT;
    out[i] = bias[r / TT];
  }
}

__global__ void zero_kernel(float4* __restrict__ p, int n4) {
  int i = blockIdx.x * blockDim.x + threadIdx.x;
  if (i < n4) p[i] = make_float4(0.f, 0.f, 0.f, 0.f);
}

// agg[tgt*CT + j] += ew * x[src*CT + j]  (fully coalesced gather + atomics)
__global__ void scatter_kernel(const float* __restrict__ x,
                               const int* __restrict__ ei,
                               const float* __restrict__ ew,
                               float* __restrict__ agg, int E) {
  int wid  = (blockIdx.x * blockDim.x + threadIdx.x) >> 5;  // one wave32 per edge
  int lane = threadIdx.x & 31;
  if (wid >= E) return;
  int src = ei[wid];
  int tgt = ei[E + wid];
  float w = ew[wid];
  const float4* xs = (const float4*)(x + (size_t)src * CT);
  float* at = agg + (size_t)tgt * CT;
  __builtin_prefetch(at, 0, 1);   // global_prefetch_b8: warm target line path
  #pragma unroll
  for (int i = 0; i < 3; ++i) {           // 3 * 32 lanes * float4 = 384 floats
    int j = i * 32 + lane;
    float4 v = xs[j];
    int b = j * 4;
    atomicAdd(at + b + 0, w * v.x);
    atomicAdd(at + b + 1, w * v.y);
    atomicAdd(at + b + 2, w * v.z);
    atomicAdd(at + b + 3, w * v.w);
  }
}

// out[n, cout, t] += sum_cin W[cin, cout] * agg[n, cin, t]
// GEMM: D(32 x ncols) = W^T (32x32) * Agg(32 x ncols), ncols = N*T, tiled 16 cols/wave.
__global__ void gemm_accum_kernel(const float* __restrict__ agg,
                                  const float* __restrict__ W,
                                  float* __restrict__ out, int ntiles) {
  int wid = blockIdx.x * (blockDim.x >> 5) + (threadIdx.x >> 5);
  if (wid >= ntiles) return;              // wave-uniform: EXEC stays all-ones for WMMA
  int lane = threadIdx.x & 31;
  int m = lane & 15;                      // A row (cout) / B,C,D column within tile
  int h = lane >> 4;                      // lane half selects K pair / M offset
  int q = wid * 16 + m;                   // global column = (node, t)
  int node = q / TT;
  int t = q - node * TT;
  const float* bcol = agg + (size_t)node * CT + t;   // + cin*TT

  v8f acc0 = {0.f,0.f,0.f,0.f,0.f,0.f,0.f,0.f};      // couts 0..15
  v8f acc1 = {0.f,0.f,0.f,0.f,0.f,0.f,0.f,0.f};      // couts 16..31
  #pragma unroll
  for (int k0 = 0; k0 < CC; k0 += 4) {
    int k = k0 + 2 * h;                   // ISA A/B layout: lane half h holds K = k0+2h, +1
    v2f b;  b[0]  = bcol[k * TT];         b[1]  = bcol[(k + 1) * TT];
    v2f a0; a0[0] = W[k * CC + m];        a0[1] = W[(k + 1) * CC + m];
    v2f a1; a1[0] = W[k * CC + 16 + m];   a1[1] = W[(k + 1) * CC + 16 + m];
    acc0 = __builtin_amdgcn_wmma_f32_16x16x4_f32(false, a0, false, b,
                                                 (short)0, acc0, false, false);
    acc1 = __builtin_amdgcn_wmma_f32_16x16x4_f32(false, a1, false, b,
                                                 (short)0, acc1, false, false);
  }
  // D layout: lane L, VGPR v -> M = v + 8*(L>>4), N = L&15
  float* orow = out + (size_t)node * CT + t;          // + cout*TT
  #pragma unroll
  for (int v = 0; v < 8; ++v) {
    int c0 = v + 8 * h;
    orow[c0 * TT]        += acc0[v];
    orow[(c0 + 16) * TT] += acc1[v];
  }
}

extern "C" void kernel_launch(void* const* d_in, const int* in_sizes, int n_in,
                              void* d_out, int out_size, void* d_ws, size_t ws_size,
                              hipStream_t stream) {
  // dict order: x, (ei,ew,W) x {fwd1,fwd2,bck1,bck2}, bias
  const float* x    = (const float*)d_in[0];
  const float* bias = (const float*)d_in[13];
  float* out = (float*)d_out;
  float* agg = (float*)d_ws;              // N*C*T floats = ~46 MB scratch

  int total  = out_size;                  // N*C*T
  int N      = total / CT;
  int E      = in_sizes[2];               // ew_fwd1 element count
  int ntiles = (N * TT) / 16;             // 360000/16 = 22500 column tiles

  init_out_kernel<<<(total + 255) / 256, 256, 0, stream>>>(out, bias, total);

  const int base_idx[4] = {1, 4, 7, 10};
  for (int a = 0; a < 4; ++a) {
    const int*   ei = (const int*)  d_in[base_idx[a] + 0];
    const float* ew = (const float*)d_in[base_idx[a] + 1];
    const float* W  = (const float*)d_in[base_idx[a] + 2];

    int n4 = total / 4;
    zero_kernel<<<(n4 + 255) / 256, 256, 0, stream>>>((float4*)agg, n4);
    scatter_kernel<<<(E + 7) / 8, 256, 0, stream>>>(x, ei, ew, agg, E);
    gemm_accum_kernel<<<(ntiles + 7) / 8, 256, 0, stream>>>(agg, W, out, ntiles);
  }
}